// scaling_classwise_training_loss_43997644980659
// MI455X (gfx1250) — compile-verified
//
#include <hip/hip_runtime.h>
#include <hip/hip_bf16.h>

#define NCLS 1000
#define NROWS_PER_BLK 8   // 8 waves of 32 per block, one wave per row

typedef float v2f __attribute__((ext_vector_type(2)));
typedef float v8f __attribute__((ext_vector_type(8)));

__device__ __forceinline__ float wave_max32(float v) {
#pragma unroll
    for (int o = 16; o > 0; o >>= 1) v = fmaxf(v, __shfl_xor(v, o, 32));
    return v;
}
__device__ __forceinline__ float wave_sum32(float v) {
#pragma unroll
    for (int o = 16; o > 0; o >>= 1) v += __shfl_xor(v, o, 32);
    return v;
}

// ---------------------------------------------------------------------------
// Phase 0: zero the class accumulators in workspace (must re-zero every call).
// ---------------------------------------------------------------------------
__global__ void ws_zero_kernel(float* __restrict__ p, int n) {
    int i = blockIdx.x * blockDim.x + threadIdx.x;
    if (i < n) p[i] = 0.0f;
}

// ---------------------------------------------------------------------------
// Phase 1: one wave32 per row. Single HBM pass: the whole 1000-float row lives
// in 8 float4 registers per lane (32 lanes x 32 floats = 1024 slots, the 24
// pad slots are -inf so exp(pad - max) == 0). Coalesced global_load_b128.
// ---------------------------------------------------------------------------
__global__ __launch_bounds__(NROWS_PER_BLK * 32)
void ce_rows_kernel(const float* __restrict__ inp, const int* __restrict__ tgt,
                    float* __restrict__ cls_sum, float* __restrict__ cls_cnt) {
    const int lane = threadIdx.x & 31;
    const int wave = threadIdx.x >> 5;
    const int row  = blockIdx.x * NROWS_PER_BLK + wave;

    const float* rowp = inp + (size_t)row * NCLS;
    const float4* rp4 = reinterpret_cast<const float4*>(rowp);

    // Uniform (same address across the wave) early loads: scalarizes nicely and
    // overlaps the pick with the reductions.
    const int   t      = tgt[row];
    const float picked = rowp[t];

    float4 v[8];
#pragma unroll
    for (int j = 0; j < 7; ++j) v[j] = rp4[lane + 32 * j];  // elements 0..895
    // elements 896..999 = 26 float4s; pad lanes 26..31 with -inf
    v[7] = (lane < 26) ? rp4[lane + 224]
                       : make_float4(-__builtin_inff(), -__builtin_inff(),
                                     -__builtin_inff(), -__builtin_inff());

    float m = -__builtin_inff();
#pragma unroll
    for (int j = 0; j < 8; ++j)
        m = fmaxf(m, fmaxf(fmaxf(v[j].x, v[j].y), fmaxf(v[j].z, v[j].w)));
    m = wave_max32(m);

    float s = 0.0f;
#pragma unroll
    for (int j = 0; j < 8; ++j) {
        s += __expf(v[j].x - m);
        s += __expf(v[j].y - m);
        s += __expf(v[j].z - m);
        s += __expf(v[j].w - m);
    }
    s = wave_sum32(s);

    if (lane == 0) {
        const float ce = m + __logf(s) - picked;   // logsumexp - picked
        atomicAdd(&cls_sum[t], ce);                // global_atomic_add_f32
        atomicAdd(&cls_cnt[t], 1.0f);
    }
}

// ---------------------------------------------------------------------------
// Phase 2: single wave32. losses[c] = sum[c]/cnt[c] into LDS (padded to 1024
// with zeros). Reductions run on the matrix pipe: V_WMMA_F32_16X16X4_F32 with
// A = ones makes every row of D the column-sums of B, so accumulating C over
// 16 chunks of 64 values reduces 1024 floats; sum of D vgpr0 over 32 lanes is
// exactly 2x the grand total (rows M=0 and M=8). Three accumulators:
//   acc1 = sum(x), acc2 = sum(x^2), acc3 = sum(x * scale(x)).
// EXEC is all-ones throughout the WMMA region (uniform 32-lane control flow).
// ---------------------------------------------------------------------------
__global__ __launch_bounds__(32)
void finalize_kernel(const float* __restrict__ cls_sum,
                     const float* __restrict__ cls_cnt,
                     float* __restrict__ out) {
    __shared__ float loss[1024];
    const int lane = threadIdx.x;

#pragma unroll
    for (int k = 0; k < 32; ++k) {
        const int c = lane + 32 * k;
        loss[c] = (c < NCLS) ? (cls_sum[c] / cls_cnt[c]) : 0.0f;
    }
    __syncthreads();

    v2f ones; ones.x = 1.0f; ones.y = 1.0f;
    v8f acc1 = {};  // sum x
    v8f acc2 = {};  // sum x^2
#pragma unroll
    for (int c = 0; c < 16; ++c) {
        v2f b;
        b.x = loss[c * 64 + 2 * lane];
        b.y = loss[c * 64 + 2 * lane + 1];
        acc1 = __builtin_amdgcn_wmma_f32_16x16x4_f32(
            false, ones, false, b, (short)0, acc1, false, false);
        v2f b2; b2.x = b.x * b.x; b2.y = b.y * b.y;
        acc2 = __builtin_amdgcn_wmma_f32_16x16x4_f32(
            false, ones, false, b2, (short)0, acc2, false, false);
    }
    const float S1 = wave_sum32(acc1[0]) * 0.5f;
    const float S2 = wave_sum32(acc2[0]) * 0.5f;

    const float mean = S1 * (1.0f / (float)NCLS);
    float var = (S2 - S1 * S1 * (1.0f / (float)NCLS)) * (1.0f / (float)(NCLS - 1));
    var = fmaxf(var, 0.0f);
    const float inv_std = rsqrtf(var);             // 1/std (ddof=1)

    // scale(x) = BETA/(1+exp(-norm/ALPHA)) - BETA/2 + 1, ALPHA=1, BETA=1.5
    v8f acc3 = {};
#pragma unroll
    for (int c = 0; c < 16; ++c) {
        const float x0 = loss[c * 64 + 2 * lane];
        const float x1 = loss[c * 64 + 2 * lane + 1];
        const float n0 = (x0 - mean) * inv_std;
        const float n1 = (x1 - mean) * inv_std;
        const float sc0 = 1.5f / (1.0f + __expf(-n0)) - 0.75f + 1.0f;
        const float sc1 = 1.5f / (1.0f + __expf(-n1)) - 0.75f + 1.0f;
        v2f w; w.x = x0 * sc0; w.y = x1 * sc1;     // pad entries are 0 -> contribute 0
        acc3 = __builtin_amdgcn_wmma_f32_16x16x4_f32(
            false, ones, false, w, (short)0, acc3, false, false);
    }
    const float result = wave_sum32(acc3[0]) * 0.5f;
    if (lane == 0) out[0] = result;
}

// ---------------------------------------------------------------------------
extern "C" void kernel_launch(void* const* d_in, const int* in_sizes, int n_in,
                              void* d_out, int out_size, void* d_ws, size_t ws_size,
                              hipStream_t stream) {
    const float* inp = (const float*)d_in[0];   // [65536, 1000] f32
    const int*   tgt = (const int*)d_in[1];     // [65536] i32
    float*       out = (float*)d_out;           // scalar f32

    float* cls_sum = (float*)d_ws;              // [1024] (padded)
    float* cls_cnt = cls_sum + 1024;            // [1024] (padded)

    const int nrows = in_sizes[1];              // 65536

    ws_zero_kernel<<<(2048 + 255) / 256, 256, 0, stream>>>(cls_sum, 2048);
    ce_rows_kernel<<<nrows / NROWS_PER_BLK, NROWS_PER_BLK * 32, 0, stream>>>(
        inp, tgt, cls_sum, cls_cnt);
    finalize_kernel<<<1, 32, 0, stream>>>(cls_sum, cls_cnt, out);
}